// TCVAE_38525856645745
// MI455X (gfx1250) — compile-verified
//
#include <hip/hip_runtime.h>

#define LOG_2PI 1.8378770664093453f
#define EXP_NEG1 0.36787944117144233f
#define NCHUNK 8

typedef __attribute__((ext_vector_type(2))) float floatx2;
typedef __attribute__((ext_vector_type(8))) float floatx8;

__device__ __forceinline__ float warp_sum32(float v) {
#pragma unroll
    for (int off = 16; off > 0; off >>= 1) v += __shfl_xor(v, off, 32);
    return v;
}

// log importance weight (rtqichen MSS): default 1/M, diag 1/N, superdiag strat,
// plus (B-1,1) and (B-2,0) strat.  (Used only in the cheap joint path.)
__device__ __forceinline__ float liw_f(int i, int j, float lM, float lN, float lS, int Bc) {
    float w = lM;
    w = (j == i) ? lN : w;
    w = (j == i + 1) ? lS : w;
    w = ((i == Bc - 1) && (j == 1)) ? lS : w;
    w = ((i == Bc - 2) && (j == 0)) ? lS : w;
    return w;
}

// ---------------- K0: recon MAE partial sums ----------------
__global__ void recon_kernel(const float* __restrict__ x, const float* __restrict__ rec,
                             float* __restrict__ part, int n) {
    __shared__ float sm[256];
    float s = 0.f;
    for (int idx = blockIdx.x * blockDim.x + threadIdx.x; idx < n; idx += gridDim.x * blockDim.x)
        s += fabsf(x[idx] - rec[idx]);
    sm[threadIdx.x] = s;
    __syncthreads();
    for (int off = 128; off > 0; off >>= 1) {
        if ((int)threadIdx.x < off) sm[threadIdx.x] += sm[threadIdx.x + off];
        __syncthreads();
    }
    if (threadIdx.x == 0) part[blockIdx.x] = sm[0];
}

// ---------------- K1: per-row prep (lane == d, D = 32) ----------------
__global__ void prep_kernel(const float* __restrict__ mu, const float* __restrict__ lv,
                            const float* __restrict__ z,
                            float* __restrict__ Amat, float* __restrict__ Bmat,
                            float* __restrict__ prarr, float* __restrict__ carr,
                            float* __restrict__ logq, float* __restrict__ logprior,
                            float lM, int B) {
    int gid = blockIdx.x * blockDim.x + threadIdx.x;
    int row = gid >> 5, d = gid & 31;
    if (row >= B) return;
    float lvv = lv[row * 32 + d];
    float muv = mu[row * 32 + d];
    float zv  = z[row * 32 + d];
    float p = __expf(-lvv);
    // GEMM operands: S[i,j] = sum_k A[i,k]*Brow[j,k] = sum_d (z^2 - 2 z mu) * p
    Amat[row * 64 + d]      = zv * zv;
    Amat[row * 64 + 32 + d] = zv;
    Bmat[row * 64 + d]      = p;
    Bmat[row * 64 + 32 + d] = -2.f * muv * p;
    // per-dim density coefficients, default log-weight lM pre-folded:
    //   v = fma(w*w, p2, r) == density + lM
    prarr[(row * 32 + d) * 2 + 0] = -0.5f * p;
    prarr[(row * 32 + d) * 2 + 1] = -0.5f * (lvv + LOG_2PI) + lM;
    // c[j] = sum_d (mu^2 p + lv + log2pi)  (joint completion term)
    float cpart = muv * muv * p + lvv + LOG_2PI;
    float w = zv - muv;
    float lqpart = -0.5f * (w * w * p + lvv + LOG_2PI);
    float lppart = -0.5f * (zv * zv * EXP_NEG1 + 1.0f + LOG_2PI);
    cpart  = warp_sum32(cpart);
    lqpart = warp_sum32(lqpart);
    lppart = warp_sum32(lppart);
    if (d == 0) {
        carr[row]     = cpart;
        logq[row]     = lqpart;
        logprior[row] = lppart;
    }
}

// ---------------- K2: pairwise — WMMA GEMM (joint) + per-dim LSE ----------------
// Block = one i-tile (16 rows); wave w = j-chunk w (B/NCHUNK columns each).
// Chunk partials merged through LDS inside the block.
__global__ void __launch_bounds__(256) pairwise_kernel(
    const float* __restrict__ Amat, const float* __restrict__ Bmat,
    const float* __restrict__ mu, const float* __restrict__ z,
    const float* __restrict__ prarr, const float* __restrict__ carr,
    const int* __restrict__ dsize,
    float* __restrict__ logqz, float* __restrict__ logqzprod,
    float lM, int B) {
    __shared__ float ldsP[NCHUNK][16][32];  // per-dim chunk partials
    __shared__ float ldsJ[NCHUNK][16][2];   // joint (m,s) chunk partials

    const int lane  = threadIdx.x & 31;
    const int chunk = threadIdx.x >> 5;     // wave id == j-chunk
    const int ibase = blockIdx.x * 16;
    const int jtiles = (B / 16) / NCHUNK;
    const int jt0 = chunk * jtiles;
    const int ln = lane & 15, hi = lane >> 4;

    const float Nf = (float)dsize[0];
    const float Mf = (float)(B - 1);
    const float lN = -__logf(Nf);
    const float lS = __logf((Nf - Mf) / (Nf * Mf));
    const float wMlin = 1.0f / Mf;
    const float wNlin = 1.0f / Nf;
    const float wSlin = (Nf - Mf) / (Nf * Mf);

    // Preload full K=64 A strip for this i-tile (ISA 32-bit A layout:
    // lane<16 holds K = 4k,4k+1; lane>=16 holds K = 4k+2,4k+3; M = ln).
    floatx2 aReg[16];
    const float* Arow = Amat + (size_t)(ibase + ln) * 64 + hi * 2;
#pragma unroll
    for (int kk = 0; kk < 16; kk++) aReg[kk] = *(const floatx2*)(Arow + kk * 4);

    float zi[16];
#pragma unroll
    for (int ii = 0; ii < 16; ii++) zi[ii] = z[(size_t)(ibase + ii) * 32 + lane];

    float m8[8], s8[8], spd[16];
#pragma unroll
    for (int r = 0; r < 8; r++) { m8[r] = -3.0e38f; s8[r] = 0.f; }
#pragma unroll
    for (int ii = 0; ii < 16; ii++) spd[ii] = 0.f;

    for (int jt = jt0; jt < jt0 + jtiles; ++jt) {
        const int jbase = jt * 16;
        // ---- joint GEMM tile: 16 x f32 WMMAs over K=64 ----
        floatx8 c = {};
        const float* Brow = Bmat + (size_t)(jbase + ln) * 64 + hi * 2;
#pragma unroll
        for (int kk = 0; kk < 16; kk++) {
            floatx2 b = *(const floatx2*)(Brow + kk * 4);
            c = __builtin_amdgcn_wmma_f32_16x16x4_f32(
                    false, aReg[kk], false, b, (short)0, c, false, false);
        }
        // joint value: logiw(i,j) - 0.5*(S + c[j]); online logsumexp over j
        float cj = carr[jbase + ln];
        int col  = jbase + ln;
        int rowb = ibase + hi * 8;
#pragma unroll
        for (int r = 0; r < 8; r++) {
            float v  = liw_f(rowb + r, col, lM, lN, lS, B) - 0.5f * (c[r] + cj);
            float nm = fmaxf(m8[r], v);
            s8[r] = s8[r] * __expf(m8[r] - nm) + __expf(v - nm);
            m8[r] = nm;
        }
        // ---- per-dim pass (lane == d), uniform weight lM pre-folded ----
#pragma unroll 4
        for (int jj = 0; jj < 16; jj++) {
            int j = jbase + jj;
            floatx2 pr = *(const floatx2*)(prarr + ((size_t)j * 32 + lane) * 2);
            float muj = mu[(size_t)j * 32 + lane];
#pragma unroll
            for (int ii = 0; ii < 16; ii++) {
                float w = zi[ii] - muj;
                float v = fmaf(w * w, pr.x, pr.y);
                spd[ii] += __expf(v);
            }
        }
    }

    // ---- per-dim weight corrections (rare special logiw entries) ----
    {
        const int j0 = jt0 * 16, j1 = j0 + jtiles * 16;
        auto addCorr = [&](int ii, int j, float wdiff) {
            floatx2 pr = *(const floatx2*)(prarr + ((size_t)j * 32 + lane) * 2);
            float muj = mu[(size_t)j * 32 + lane];
            float w = zi[ii] - muj;
            float vd = fmaf(w * w, pr.x, pr.y) - lM;  // raw density
            spd[ii] += __expf(vd) * wdiff;
        };
#pragma unroll 1
        for (int ii = 0; ii < 16; ii++) {
            int i = ibase + ii;
            if (i >= j0 && i < j1)                           addCorr(ii, i,     wNlin - wMlin);
            if (i + 1 < B && i + 1 >= j0 && i + 1 < j1)      addCorr(ii, i + 1, wSlin - wMlin);
            if (i == B - 1 && 1 >= j0 && 1 < j1)             addCorr(ii, 1,     wSlin - wMlin);
            if (i == B - 2 && 0 >= j0 && 0 < j1)             addCorr(ii, 0,     wSlin - wMlin);
        }
    }

    // joint: merge (m,s) across the 16 lanes of each half (cols differ per lane)
#pragma unroll
    for (int r = 0; r < 8; r++) {
        float m = m8[r], s = s8[r];
#pragma unroll
        for (int off = 1; off < 16; off <<= 1) {
            float om = __shfl_xor(m, off, 16);
            float os = __shfl_xor(s, off, 16);
            float nm = fmaxf(m, om);
            s = s * __expf(m - nm) + os * __expf(om - nm);
            m = nm;
        }
        if (ln == 0) {
            ldsJ[chunk][hi * 8 + r][0] = m;
            ldsJ[chunk][hi * 8 + r][1] = s;
        }
    }
    // per-dim chunk partials -> LDS
#pragma unroll
    for (int ii = 0; ii < 16; ii++) ldsP[chunk][ii][lane] = spd[ii];

    __syncthreads();

    // ---- in-block cross-chunk merge (wave 0) ----
    if (chunk == 0) {
        if (lane < 16) {  // one row per lane
            float m = -3.0e38f, s = 0.f;
#pragma unroll
            for (int cc = 0; cc < NCHUNK; cc++) {
                float pm = ldsJ[cc][lane][0], ps = ldsJ[cc][lane][1];
                float nm = fmaxf(m, pm);
                s = s * __expf(m - nm) + ps * __expf(pm - nm);
                m = nm;
            }
            logqz[ibase + lane] = m + __logf(s);
        }
#pragma unroll 1
        for (int ii = 0; ii < 16; ii++) {
            float t = 0.f;
#pragma unroll
            for (int cc = 0; cc < NCHUNK; cc++) t += ldsP[cc][ii][lane];
            float lt = __logf(t);
            lt = warp_sum32(lt);
            if (lane == 0) logqzprod[ibase + ii] = lt;
        }
    }
}

// ---------------- K3: final scalar reduction ----------------
__global__ void final_kernel(const float* __restrict__ part, int nparts,
                             const float* __restrict__ logq, const float* __restrict__ logqz,
                             const float* __restrict__ logqzprod, const float* __restrict__ logprior,
                             int B, int n_recon, float* __restrict__ out) {
    __shared__ float sm[256 * 4];
    float sr = 0, smi = 0, stc = 0, sdw = 0;
    for (int i = threadIdx.x; i < nparts; i += 256) sr += part[i];
    for (int i = threadIdx.x; i < B; i += 256) {
        smi += logq[i] - logqz[i];
        stc += logqz[i] - logqzprod[i];
        sdw += logqzprod[i] - logprior[i];
    }
    sm[threadIdx.x] = sr;
    sm[256 + threadIdx.x] = smi;
    sm[512 + threadIdx.x] = stc;
    sm[768 + threadIdx.x] = sdw;
    __syncthreads();
    for (int off = 128; off > 0; off >>= 1) {
        if ((int)threadIdx.x < off) {
            sm[threadIdx.x] += sm[threadIdx.x + off];
            sm[256 + threadIdx.x] += sm[256 + threadIdx.x + off];
            sm[512 + threadIdx.x] += sm[512 + threadIdx.x + off];
            sm[768 + threadIdx.x] += sm[768 + threadIdx.x + off];
        }
        __syncthreads();
    }
    if (threadIdx.x == 0) {
        float recon = sm[0] / (float)n_recon;
        float Bf = (float)B;
        // ALPHA = BETA = GAMMA = 1
        float kl = 1.0f * (sm[256] / Bf) + 1.0f * (sm[512] / Bf) + 1.0f * (sm[768] / Bf);
        out[0] = recon + kl;
        out[1] = recon;
        out[2] = kl;
    }
}

extern "C" void kernel_launch(void* const* d_in, const int* in_sizes, int n_in,
                              void* d_out, int out_size, void* d_ws, size_t ws_size,
                              hipStream_t stream) {
    const float* rec = (const float*)d_in[0];
    const float* x   = (const float*)d_in[1];
    const float* mu  = (const float*)d_in[2];
    const float* lv  = (const float*)d_in[3];
    const float* z   = (const float*)d_in[4];
    const int*   ds  = (const int*)d_in[5];
    float* out = (float*)d_out;

    const int D = 32;
    const int B = in_sizes[2] / D;   // 2048
    const int n_recon = in_sizes[0]; // B*T
    const float lM = -logf((float)(B - 1));

    float* ws = (float*)d_ws;
    float* Amat      = ws; ws += (size_t)B * 64;
    float* Bmat      = ws; ws += (size_t)B * 64;
    float* prarr     = ws; ws += (size_t)B * 64;
    float* carr      = ws; ws += B;
    float* logq      = ws; ws += B;
    float* logprior  = ws; ws += B;
    float* logqz     = ws; ws += B;
    float* logqzprod = ws; ws += B;
    float* reconPart = ws; ws += 256;

    recon_kernel<<<256, 256, 0, stream>>>(x, rec, reconPart, n_recon);
    prep_kernel<<<(B * 32 + 255) / 256, 256, 0, stream>>>(mu, lv, z, Amat, Bmat, prarr,
                                                          carr, logq, logprior, lM, B);
    pairwise_kernel<<<B / 16, 256, 0, stream>>>(Amat, Bmat, mu, z, prarr, carr, ds,
                                                logqz, logqzprod, lM, B);
    final_kernel<<<1, 256, 0, stream>>>(reconPart, 256, logq, logqz, logqzprod, logprior,
                                        B, n_recon, out);
}